// DepthwiseConv4d_8220567404796
// MI455X (gfx1250) — compile-verified
//
#include <hip/hip_runtime.h>
#include <cstdint>

typedef float v2f __attribute__((ext_vector_type(2)));
typedef float v4f __attribute__((ext_vector_type(4)));

#define XT 4
#define YT 8
#define XH (XT + 2)          // 6
#define YH (YT + 2)          // 10
#define NLINES (XH * YH)     // 60 (x,y) lines
#define ZSTRIDE 484          // dwords per z-row: 60*8 + 4 pad (16B aligned, gcd(484,64)=4)
#define LDS_FLOATS (32 * ZSTRIDE)      // 15488 floats = 61952 B
#define CHUNKS (NLINES * 64)           // 3840 16B chunks staged
#define THREADS 256
#define CHUNKS_PER_THREAD (CHUNKS / THREADS)  // 15

__global__ __launch_bounds__(THREADS)
void dwconv4d_kernel(const float* __restrict__ in,
                     const float* __restrict__ kg,
                     float* __restrict__ out)
{
    // z-major tile: lines[z][line][t] with padded z-row stride.
    __shared__ __align__(16) float lines[LDS_FLOATS];
    __shared__ float wsm[81];

    const int tid    = threadIdx.x;
    const int tileid = blockIdx.x;          // 0..31
    const int x0     = (tileid >> 2) * XT;  // 8 x-tiles of 4
    const int y0     = (tileid & 3) * YT;   // 4 y-tiles of 8
    const int c      = blockIdx.y;          // 64 channels
    const int b      = blockIdx.z;          // 4 batches

    const size_t gbase = ((size_t)(b * 64 + c)) << 18;   // * 32*32*32*8
    const float* gin   = in + gbase;

    // ---- Stage tile: async global->LDS DMA (ASYNCcnt), scattered z-major ----
    const uint32_t lds_base = (uint32_t)(uintptr_t)(&lines[0]);

    #pragma unroll
    for (int k = 0; k < CHUNKS_PER_THREAD; ++k) {
        const int cid  = k * THREADS + tid;
        const int line = cid >> 6;          // which (x,y) line (0..59)
        const int c16  = cid & 63;          // 16B chunk within the 1KB line
        const int zz   = c16 >> 1;          // z slot 0..31
        const int half = c16 & 1;           // low/high 16B of the 8-float z-group
        const int lx = line / YH, ly = line % YH;
        const int gx = x0 - 1 + lx, gy = y0 - 1 + ly;
        const uint32_t ldsoff = (uint32_t)(zz * ZSTRIDE + line * 8 + half * 4) * 4u;
        if ((unsigned)gx < 32u && (unsigned)gy < 32u) {
            const float* gsrc = gin + (((size_t)(gx * 32 + gy)) << 8) + c16 * 4;
            const uint32_t ldst = lds_base + ldsoff;
            asm volatile("global_load_async_to_lds_b128 %0, %1, off"
                         :: "v"(ldst), "v"(gsrc) : "memory");
        } else {
            *(v4f*)(&lines[zz * ZSTRIDE + line * 8 + half * 4]) = (v4f)(0.f);
        }
    }
    if (tid < 81) wsm[tid] = kg[tid * 64 + c];   // per-channel tap weights

#if __has_builtin(__builtin_amdgcn_s_wait_asynccnt)
    __builtin_amdgcn_s_wait_asynccnt(0);
#else
    asm volatile("s_wait_asynccnt 0" ::: "memory");
#endif
    __syncthreads();

    // ---- Compute: one z per thread, two y-adjacent point-pairs, T in regs.
    // Pair data {l0[t], l1[t]} is 8 dwords apart -> ds_load_2addr_b32 into a
    // consecutive VGPR pair -> feeds v_pk_fma_f32 with no shuffle movs.
    const int z = tid & 31;
    const int q = tid >> 5;   // 0..7

    for (int pr = 0; pr < 2; ++pr) {
        const int pairid = q + pr * 8;      // 0..15
        const int pt0    = pairid * 2;
        const int xl     = pt0 >> 3;        // 0..3
        const int yl     = pt0 & 7;         // 0,2,4,6 ; partner at yl+1

        v2f A[8];
        #pragma unroll
        for (int t = 0; t < 8; ++t) A[t] = (v2f)(0.f);

        for (int dxy = 0; dxy < 9; ++dxy) {          // 9 (dx,dy) taps, rolled
            const int dx = dxy / 3, dy = dxy % 3;
            const int lbase = ((xl + dx) * YH + (yl + dy)) * 8;

            #pragma unroll
            for (int dz = 0; dz < 3; ++dz) {
                int zp = z + dz - 1;
                const float msk = ((unsigned)zp < 32u) ? 1.f : 0.f;  // z 'same' pad
                zp = (zp < 0) ? 0 : (zp > 31 ? 31 : zp);
                const int wb = dxy * 9 + dz * 3;     // tap = ((dx*3+dy)*3+dz)*3+dt
                const v2f w0 = (v2f)(wsm[wb + 0] * msk);
                const v2f w1 = (v2f)(wsm[wb + 1] * msk);
                const v2f w2 = (v2f)(wsm[wb + 2] * msk);

                const float* p = &lines[zp * ZSTRIDE + lbase];
                v2f V[8];
                #pragma unroll
                for (int t = 0; t < 8; ++t) { V[t].x = p[t]; V[t].y = p[t + 8]; }

                #pragma unroll
                for (int t = 0; t < 8; ++t) A[t] += w1 * V[t];        // dt=1
                #pragma unroll
                for (int t = 1; t < 8; ++t) A[t] += w0 * V[t - 1];    // dt=0 (t'=-1 pad)
                #pragma unroll
                for (int t = 0; t < 7; ++t) A[t] += w2 * V[t + 1];    // dt=2 (t'=8 pad)
            }
        }

        // ---- Store: 8 contiguous T floats per point, non-temporal (write-once,
        // never re-read: keep L2 free for the input halo working set) ----
        v4f* o0 = (v4f*)(out + gbase +
                  (((size_t)((x0 + xl) * 32 + (y0 + yl))) << 8) + z * 8);
        v4f* o1 = o0 + 64;   // y+1 point (+256 floats)
        v4f p0lo = (v4f){A[0].x, A[1].x, A[2].x, A[3].x};
        v4f p0hi = (v4f){A[4].x, A[5].x, A[6].x, A[7].x};
        v4f p1lo = (v4f){A[0].y, A[1].y, A[2].y, A[3].y};
        v4f p1hi = (v4f){A[4].y, A[5].y, A[6].y, A[7].y};
        __builtin_nontemporal_store(p0lo, o0 + 0);
        __builtin_nontemporal_store(p0hi, o0 + 1);
        __builtin_nontemporal_store(p1lo, o1 + 0);
        __builtin_nontemporal_store(p1hi, o1 + 1);
    }
}

extern "C" void kernel_launch(void* const* d_in, const int* in_sizes, int n_in,
                              void* d_out, int out_size, void* d_ws, size_t ws_size,
                              hipStream_t stream) {
    const float* in  = (const float*)d_in[0];   // [4,64,32,32,32,8] f32
    const float* kg  = (const float*)d_in[1];   // [81,64,1] f32
    float* out       = (float*)d_out;           // [4,64,32,32,32,8] f32

    dim3 grid(32, 64, 4);   // (x-tiles*y-tiles, C, B)
    dim3 block(THREADS);
    hipLaunchKernelGGL(dwconv4d_kernel, grid, block, 0, stream, in, kg, out);
}